// GIN_22428319220138
// MI455X (gfx1250) — compile-verified
//
#include <hip/hip_runtime.h>

// GIN forward for MI455X (gfx1250, wave32).
// Aggregation (dominant cost) = L2-resident f32 atomics; dense layers use
// V_WMMA_F32_16X16X4_F32 to keep full f32 precision on the matrix unit.

typedef __attribute__((ext_vector_type(2))) float v2f;
typedef __attribute__((ext_vector_type(8))) float v8f;

#define GIN_D_IN 64
#define GIN_D_HID 32
#define GIN_N_CLASSES 40
#define GIN_SELF_W 1.1f   // (1 + epsilon), epsilon = 0.1

// ---------------- agg = (1+eps) * h  (vectorized elementwise) ----------------
__global__ __launch_bounds__(256) void gin_scale_kernel(
    const float* __restrict__ h, float* __restrict__ agg, int n4) {
  int i = blockIdx.x * blockDim.x + threadIdx.x;
  if (i < n4) {
    const float4* hv = (const float4*)h;
    float4* av = (float4*)agg;
    float4 v = hv[i];
    v.x *= GIN_SELF_W; v.y *= GIN_SELF_W; v.z *= GIN_SELF_W; v.w *= GIN_SELF_W;
    av[i] = v;
  }
}

// ---------------- agg[dst] += h[src] (scatter-add over edges) ----------------
// D/4 lanes per edge; each lane does one coalesced float4 gather + 4 f32 atomics.
// Feature matrices fit in the 192MB L2 so both gather and atomic RMW stay on-chip.
template <int D>
__global__ __launch_bounds__(256) void gin_scatter_kernel(
    const float* __restrict__ h, const int* __restrict__ src,
    const int* __restrict__ dst, float* __restrict__ agg, int n_edges) {
  constexpr int TPE = D / 4;  // power of two (16 or 8)
  int gid = blockIdx.x * blockDim.x + threadIdx.x;
  if (gid >= n_edges * TPE) return;
  int e = gid / TPE;
  int f = (gid % TPE) * 4;
  int s = src[e];
  int d = dst[e];
  const float4 v = *(const float4*)(h + (size_t)s * D + f);
  float* p = agg + (size_t)d * D + f;
  atomicAdd(p + 0, v.x);
  atomicAdd(p + 1, v.y);
  atomicAdd(p + 2, v.z);
  atomicAdd(p + 3, v.w);
}

// ---------------- C = [relu](A @ W) via V_WMMA_F32_16X16X4_F32 ----------------
// One wave per 16-row tile; NT col-tiles of 16; K consumed 4 at a time.
// A frag (2 VGPR/lane): lane L holds row M=L%16, K = (L/16)*2 + {0,1}.
// B frag mirrors: lane L holds col N=L%16, K = (L/16)*2 + {0,1}.
// C/D (8 VGPR): VGPR r = row r + 8*(L/16), col L%16.
template <int DINL, int DOUTL, int NT, bool RELU>
__global__ __launch_bounds__(256) void gin_gemm_kernel(
    const float* __restrict__ A, const float* __restrict__ W,
    float* __restrict__ C, int n_rows) {
  const int lane = threadIdx.x & 31;
  const int wave = threadIdx.x >> 5;
  const int ntiles = (n_rows + 15) >> 4;
  const int tile = blockIdx.x * (blockDim.x >> 5) + wave;
  if (tile >= ntiles) return;  // wave-uniform: EXEC stays all-1 for live waves

  const int row0 = tile << 4;
  const int l16 = lane & 15;
  const int khalf = lane >> 4;  // 0 or 1
  int arow = row0 + l16;
  if (arow >= n_rows) arow = n_rows - 1;  // clamp loads; stores are guarded

  v8f acc[NT] = {};

#pragma unroll
  for (int kk = 0; kk < DINL / 4; ++kk) {
    const int kb = kk * 4 + khalf * 2;
    v2f a;
    a.x = A[(size_t)arow * DINL + kb];
    a.y = A[(size_t)arow * DINL + kb + 1];
#pragma unroll
    for (int t = 0; t < NT; ++t) {
      const int col = t * 16 + l16;
      const bool ok = ((DOUTL & 15) == 0) || (col < DOUTL);
      v2f b;
      b.x = ok ? W[(size_t)kb * DOUTL + col] : 0.0f;
      b.y = ok ? W[(size_t)(kb + 1) * DOUTL + col] : 0.0f;
      acc[t] = __builtin_amdgcn_wmma_f32_16x16x4_f32(
          false, a, false, b, (short)0, acc[t], false, false);
    }
  }

#pragma unroll
  for (int t = 0; t < NT; ++t) {
    const int col = t * 16 + l16;
    if (((DOUTL & 15) != 0) && col >= DOUTL) continue;
#pragma unroll
    for (int r = 0; r < 8; ++r) {
      const int row = row0 + r + khalf * 8;
      if (row < n_rows) {
        float v = acc[t][r];
        if (RELU) v = fmaxf(v, 0.0f);
        C[(size_t)row * DOUTL + col] = v;
      }
    }
  }
}

extern "C" void kernel_launch(void* const* d_in, const int* in_sizes, int n_in,
                              void* d_out, int out_size, void* d_ws, size_t ws_size,
                              hipStream_t stream) {
  const float* x  = (const float*)d_in[0];
  const float* W0 = (const float*)d_in[1];
  const float* W1 = (const float*)d_in[2];
  const float* W2 = (const float*)d_in[3];
  const int* src  = (const int*)d_in[4];
  const int* dst  = (const int*)d_in[5];
  float* out = (float*)d_out;

  const int n = in_sizes[0] / GIN_D_IN;  // 100000 nodes
  const int n_edges = in_sizes[4];       // 1.6M edges

  // Workspace: agg buffer (n*64 f32, reused for all layers) + h buffer (n*32 f32)
  float* agg = (float*)d_ws;
  float* hbuf = agg + (size_t)n * GIN_D_IN;

  const int ntiles = (n + 15) / 16;
  const int gemm_blocks = (ntiles + 7) / 8;  // 8 waves/block, 1 tile/wave

  // ---- Layer 1: h1 = relu((A @ x_agg) @ W0), D_in=64 -> 32 ----
  {
    int n4 = n * GIN_D_IN / 4;
    gin_scale_kernel<<<(n4 + 255) / 256, 256, 0, stream>>>(x, agg, n4);
    int tot = n_edges * (GIN_D_IN / 4);
    gin_scatter_kernel<GIN_D_IN>
        <<<(tot + 255) / 256, 256, 0, stream>>>(x, src, dst, agg, n_edges);
    gin_gemm_kernel<GIN_D_IN, GIN_D_HID, 2, true>
        <<<gemm_blocks, 256, 0, stream>>>(agg, W0, hbuf, n);
  }
  // ---- Layer 2: h2 = relu(agg(h1) @ W1), 32 -> 32 ----
  {
    int n4 = n * GIN_D_HID / 4;
    gin_scale_kernel<<<(n4 + 255) / 256, 256, 0, stream>>>(hbuf, agg, n4);
    int tot = n_edges * (GIN_D_HID / 4);
    gin_scatter_kernel<GIN_D_HID>
        <<<(tot + 255) / 256, 256, 0, stream>>>(hbuf, src, dst, agg, n_edges);
    gin_gemm_kernel<GIN_D_HID, GIN_D_HID, 2, true>
        <<<gemm_blocks, 256, 0, stream>>>(agg, W1, hbuf, n);
  }
  // ---- Layer 3: out = agg(h2) @ W2, 32 -> 40 (3rd col-tile masked) ----
  {
    int n4 = n * GIN_D_HID / 4;
    gin_scale_kernel<<<(n4 + 255) / 256, 256, 0, stream>>>(hbuf, agg, n4);
    int tot = n_edges * (GIN_D_HID / 4);
    gin_scatter_kernel<GIN_D_HID>
        <<<(tot + 255) / 256, 256, 0, stream>>>(hbuf, src, dst, agg, n_edges);
    gin_gemm_kernel<GIN_D_HID, GIN_N_CLASSES, 3, false>
        <<<gemm_blocks, 256, 0, stream>>>(agg, W2, out, n);
  }
}